// PatchEmbed_HS_mae_13073880449624
// MI455X (gfx1250) — compile-verified
//
#include <hip/hip_runtime.h>
#include <hip/hip_bf16.h>

typedef __attribute__((ext_vector_type(2))) float v2f;
typedef __attribute__((ext_vector_type(8))) float v8f;

#define B_ 8
#define T_ 32
#define N_ 1024
#define E_ 384

// Stage A: g2[b][t][k] = bias[xs[k]] + sum_c2 w[xs[k],c2] * x[b, t%4, c2*8+t/4, ys[ys[k]], xs[ys[k]]]
__global__ __launch_bounds__(256) void stageA_gather_proj(
    const float* __restrict__ x, const float* __restrict__ w,
    const float* __restrict__ bias, const int* __restrict__ curve,
    float* __restrict__ g2) {
  int idx = blockIdx.x * blockDim.x + threadIdx.x;  // [0, 8*32*1024)
  int k = idx & 1023;
  int t = (idx >> 10) & 31;
  int b = idx >> 15;

  int e1 = curve[2 * k];        // xs[k]  in [0,32)
  int n2 = curve[2 * k + 1];    // ys[k]  in [0,32)
  int wc = curve[2 * n2];       // xs[n2]
  int hc = curve[2 * n2 + 1];   // ys[n2]

  int c  = t & 3;
  int tq = t >> 2;
  int spatial = hc * 32 + wc;
  const float* xb = x + ((size_t)(b * 4 + c) * 32) * 1024;  // x[b][c][.][.][.]

  float acc = bias[e1];
#pragma unroll
  for (int c2 = 0; c2 < 4; ++c2) {
    float xv = xb[(size_t)(c2 * 8 + tq) * 1024 + spatial];
    acc = fmaf(w[e1 * 4 + c2], xv, acc);
  }
  g2[idx] = acc;
}

// Stage B: out[b][u][k][e] = bias[e] + sum_c3 w[e,c3] * g2[b][c3*8+u][k]
// One wave handles 16 tokens (k0..k0+15) x all 384 e-columns via 24 WMMA 16x16x4 f32 tiles.
// A (16x4 f32, 2 VGPRs): lane L (M=L%16): V0 = A[M, L<16?0:2], V1 = A[M, L<16?1:3]
// B (4x16 f32, 2 VGPRs): lane L (N=L%16): V0 = B[L<16?0:2, N], V1 = B[L<16?1:3, N]
// C/D (16x16 f32, 8 VGPRs): lane L, VGPR v: D[M = v + 8*(L>=16), N = L%16]
__global__ __launch_bounds__(256) void stageB_wmma_proj(
    const float* __restrict__ g2, const float* __restrict__ w,
    const float* __restrict__ bias, float* __restrict__ out) {
  int tid  = blockIdx.x * blockDim.x + threadIdx.x;
  int lane = tid & 31;
  int wid  = tid >> 5;          // [0, 4096)
  int kt   = wid & 63;          // k-tile within (b,u)
  int u    = (wid >> 6) & 7;
  int b    = wid >> 9;

  int half = lane >> 4;         // 0: lanes 0-15, 1: lanes 16-31
  int m    = lane & 15;
  int k0   = kt * 16;
  int c3b  = half * 2;          // K-pair base for this half-wave
  int token = k0 + m;

  const float* g2b = g2 + (size_t)b * (T_ * N_);
  v2f a;
  a.x = g2b[(size_t)((c3b + 0) * 8 + u) * N_ + token];
  a.y = g2b[(size_t)((c3b + 1) * 8 + u) * N_ + token];

  float* outb = out + ((size_t)(b * 8 + u) * N_) * E_;
  int rowbase = k0 + half * 8;

#pragma unroll 4
  for (int et = 0; et < 24; ++et) {
    int col = et * 16 + m;
    v2f bm;
    bm.x = w[col * 4 + c3b];
    bm.y = w[col * 4 + c3b + 1];
    float cv = bias[col];
    v8f c = {cv, cv, cv, cv, cv, cv, cv, cv};
    v8f d = __builtin_amdgcn_wmma_f32_16x16x4_f32(
        /*neg_a=*/false, a, /*neg_b=*/false, bm,
        /*c_mod=*/(short)0, c, /*reuse_a=*/false, /*reuse_b=*/false);
#pragma unroll
    for (int v = 0; v < 8; ++v) {
      // out is write-once streaming data: non-temporal store keeps L2 for g2/w/bias
      __builtin_nontemporal_store(d[v], &outb[(size_t)(rowbase + v) * E_ + col]);
    }
  }
}

extern "C" void kernel_launch(void* const* d_in, const int* in_sizes, int n_in,
                              void* d_out, int out_size, void* d_ws, size_t ws_size,
                              hipStream_t stream) {
  const float* x     = (const float*)d_in[0];
  const float* w     = (const float*)d_in[1];
  const float* bias  = (const float*)d_in[2];
  const int*   curve = (const int*)d_in[3];
  float* out = (float*)d_out;
  float* g2  = (float*)d_ws;   // 8*32*1024 floats = 1 MB

  // Stage A: 262144 threads
  stageA_gather_proj<<<1024, 256, 0, stream>>>(x, w, bias, curve, g2);
  // Stage B: 4096 waves * 32 lanes = 131072 threads
  stageB_wmma_proj<<<512, 256, 0, stream>>>(g2, w, bias, out);
}